// STARFeatureExtractor_62216896250524
// MI455X (gfx1250) — compile-verified
//
#include <hip/hip_runtime.h>
#include <hip/hip_bf16.h>

typedef __attribute__((ext_vector_type(2))) float v2f;
typedef __attribute__((ext_vector_type(8))) float v8f;

#define TT 1024          // time length (n)
#define MM 1024          // columns = B*C
#define CC 16
#define BB 64
#define NN (5 * MM)      // packed RHS width
#define PERIOD 12
#define KK_LOWESS 614    // int(0.6 * 1024)

#define KC 32            // GEMM K-chunk staged in LDS (8 K-steps of 4)
// LDS B layout (float2 units): frag(kstep, hi, col) at kstep*160 + hi*80 + col
// hi stride = 160 floats == 32 mod 64 -> lanes 0-15 and 16-31 use disjoint banks
#define KSTEP_F2 160     // float2 per kstep block
#define HI_F2    80      // float2 per hi half

// ---------------------------------------------------------------------------
// Build tricube LOWESS weight matrix w[i][j] (data independent, closed-form h)
// ---------------------------------------------------------------------------
__global__ __launch_bounds__(256) void build_w_kernel(float* __restrict__ w) {
  int idx = blockIdx.x * blockDim.x + threadIdx.x;
  if (idx >= TT * TT) return;
  int i = idx / TT, j = idx % TT;
  int a = min(i, TT - 1 - i);
  int r1 = KK_LOWESS / 2;                       // ceil((k-1)/2)
  int h = (r1 <= a) ? r1 : (KK_LOWESS - 1 - a); // k-th smallest |i-j|
  float d = fabsf((float)(i - j));
  float u = fminf(d / fmaxf((float)h, 1e-12f), 1.0f);
  float u3 = u * u * u;
  float om = 1.0f - u3;
  w[idx] = om * om * om;
}

// (B,T,C) -> (T, M) transpose, and delta = 1
__global__ __launch_bounds__(256) void transpose_init_kernel(
    const float* __restrict__ in, float* __restrict__ y, float* __restrict__ delta) {
  int idx = blockIdx.x * blockDim.x + threadIdx.x;
  if (idx >= BB * TT * CC) return;
  int c = idx % CC;
  int t = (idx / CC) % TT;
  int b = idx / (CC * TT);
  int ym = t * MM + b * CC + c;
  y[ym] = in[idx];
  delta[ym] = 1.0f;
}

// Pack 5 weighted RHS column blocks: [delta | t*delta | t^2*delta | y*delta | t*y*delta]
__global__ __launch_bounds__(256) void build_r5_kernel(
    const float* __restrict__ y, const float* __restrict__ delta, float* __restrict__ r5) {
  int idx = blockIdx.x * blockDim.x + threadIdx.x;
  if (idx >= TT * MM) return;
  int t = idx / MM, m = idx % MM;
  float dv = delta[idx];
  float yv = y[idx];
  float tf = (float)t;
  float* row = r5 + (size_t)t * NN;
  row[m]          = dv;
  row[MM + m]     = tf * dv;
  row[2 * MM + m] = tf * tf * dv;
  row[3 * MM + m] = dv * yv;
  row[4 * MM + m] = tf * dv * yv;
}

// ---------------------------------------------------------------------------
// FP32 WMMA GEMM: C(MxN) = A(MxK) * B(KxN), row-major, M%128==0, N%64==0,
// K%KC==0. 8 waves per block; each wave computes a 16x64 strip (4 tiles).
// B chunks are double-buffered in LDS in pre-paired fragment layout so each
// lane's fragment is one aligned ds_load_b64; fragments are register-pipelined
// two K-steps deep; A fragments are register-pipelined one chunk ahead.
// ---------------------------------------------------------------------------
__global__ __launch_bounds__(256) void wmma_gemm_f32_kernel(
    const float* __restrict__ A, const float* __restrict__ B, float* __restrict__ C,
    int M, int N, int K) {
  __shared__ v2f Bs[2][8 * KSTEP_F2];   // 2 x 10240 B

  const int tid  = threadIdx.x;
  const int lane = tid & 31;
  const int wave = tid >> 5;
  const int row0 = blockIdx.y * 128 + wave * 16;
  const int col0 = blockIdx.x * 64;
  const int hi   = lane >> 4;   // 0: lanes 0-15, 1: lanes 16-31
  const int lo   = lane & 15;
  const int kOff = hi * 2;      // K-pair offset within the 4-wide K step
  const int hb   = hi * HI_F2 + lo;   // float2 base for this lane's fragments

  // B staging: 256 threads move KC*64 = 2048 floats, 8 per thread.
  // thread covers rows srow + 4*i (kstep = i, row-in-step j = srow), col scol
  const int srow = tid >> 6;    // 0..3
  const int scol = tid & 63;    // 0..63
  // float index inside a buffer for (kstep i): i*320 + (j>>1)*160 + col*2 + (j&1)
  const int wbase = ((srow >> 1) * 160) + (scol * 2) + (srow & 1);

  v8f acc0 = {}, acc1 = {}, acc2 = {}, acc3 = {};
  const float* aRowPtr = A + (size_t)(row0 + lo) * K + kOff;
  const float* bStagePtr = B + (size_t)srow * N + col0 + scol;

  // ---- stage chunk 0 ----
  float breg[8];
#pragma unroll
  for (int i = 0; i < 8; ++i) breg[i] = bStagePtr[(size_t)i * 4 * N];
  v2f a_cur[8];
#pragma unroll
  for (int kk = 0; kk < 8; ++kk) a_cur[kk] = *(const v2f*)(aRowPtr + kk * 4);
  {
    float* bsf = (float*)&Bs[0][0];
#pragma unroll
    for (int i = 0; i < 8; ++i) bsf[i * 320 + wbase] = breg[i];
  }
  __syncthreads();

#define LOADB(dst, kkc)                                      \
  {                                                          \
    dst[0] = Bs[buf][(kkc) * KSTEP_F2 + hb + 0];             \
    dst[1] = Bs[buf][(kkc) * KSTEP_F2 + hb + 16];            \
    dst[2] = Bs[buf][(kkc) * KSTEP_F2 + hb + 32];            \
    dst[3] = Bs[buf][(kkc) * KSTEP_F2 + hb + 48];            \
  }
#define WMMA4(af, bf)                                                                               \
  {                                                                                                 \
    acc0 = __builtin_amdgcn_wmma_f32_16x16x4_f32(false, af, false, bf[0], (short)0, acc0, false, false); \
    acc1 = __builtin_amdgcn_wmma_f32_16x16x4_f32(false, af, false, bf[1], (short)0, acc1, false, false); \
    acc2 = __builtin_amdgcn_wmma_f32_16x16x4_f32(false, af, false, bf[2], (short)0, acc2, false, false); \
    acc3 = __builtin_amdgcn_wmma_f32_16x16x4_f32(false, af, false, bf[3], (short)0, acc3, false, false); \
  }

  const int NC = K / KC;
  for (int c = 0; c < NC; ++c) {
    const int buf = c & 1;

    // prefetch global data for chunk c+1 while chunk c computes
    v2f a_nxt[8];
    if (c + 1 < NC) {
      const int kb = (c + 1) * KC;
#pragma unroll
      for (int i = 0; i < 8; ++i)
        breg[i] = bStagePtr[(size_t)(kb + i * 4) * N];
#pragma unroll
      for (int kk = 0; kk < 8; ++kk)
        a_nxt[kk] = *(const v2f*)(aRowPtr + kb + kk * 4);
    }
    if (c + 2 < NC) {
      // warm L2/WGP$ for chunk c+2 (global_prefetch_b8)
      __builtin_prefetch(&bStagePtr[(size_t)(c + 2) * KC * N], 0, 3);
    }

    // ---- compute chunk c from LDS, B fragments pipelined 2 K-steps deep ----
    v2f b0[4], b1[4];
    LOADB(b0, 0);
    LOADB(b1, 1);
#pragma unroll
    for (int kp = 0; kp < 4; ++kp) {
      WMMA4(a_cur[2 * kp], b0);
      if (2 * kp + 2 < 8) LOADB(b0, 2 * kp + 2);
      WMMA4(a_cur[2 * kp + 1], b1);
      if (2 * kp + 3 < 8) LOADB(b1, 2 * kp + 3);
    }

    if (c + 1 < NC) {
      __syncthreads();
      {
        float* bsf = (float*)&Bs[buf ^ 1][0];
#pragma unroll
        for (int i = 0; i < 8; ++i) bsf[i * 320 + wbase] = breg[i];
      }
      __syncthreads();
#pragma unroll
      for (int kk = 0; kk < 8; ++kk) a_cur[kk] = a_nxt[kk];
    }
  }
#undef LOADB
#undef WMMA4

  // D layout: VGPR v -> row (v + hi*8), col lo
  float* cbase = C + (size_t)(row0 + hi * 8) * N + col0 + lo;
#pragma unroll
  for (int v = 0; v < 8; ++v) {
    cbase[(size_t)v * N + 0]  = acc0[v];
    cbase[(size_t)v * N + 16] = acc1[v];
    cbase[(size_t)v * N + 32] = acc2[v];
    cbase[(size_t)v * N + 48] = acc3[v];
  }
}

// yhat from the 5 packed sums
__global__ __launch_bounds__(256) void compute_yhat_kernel(
    const float* __restrict__ s5, float* __restrict__ yhat) {
  int idx = blockIdx.x * blockDim.x + threadIdx.x;
  if (idx >= TT * MM) return;
  int t = idx / MM, m = idx % MM;
  const float* row = s5 + (size_t)t * NN;
  float S0 = row[m], S1 = row[MM + m], S2 = row[2 * MM + m];
  float Sy = row[3 * MM + m], Sxy = row[4 * MM + m];
  float mx = S1 / S0;
  float my = Sy / S0;
  float den = S2 - mx * S1;
  bool small = fabsf(den) < 1e-10f;
  float b1 = small ? 0.0f : (Sxy - mx * Sy) / den;
  yhat[idx] = my + b1 * ((float)t - mx);
}

__global__ __launch_bounds__(256) void abs_err_kernel(
    const float* __restrict__ y, const float* __restrict__ yhat, float* __restrict__ tmp) {
  int idx = blockIdx.x * blockDim.x + threadIdx.x;
  if (idx >= TT * MM) return;
  tmp[idx] = fabsf(y[idx] - yhat[idx]);
}

// ---------------------------------------------------------------------------
// Per-column selection via LDS bitonic sort of 1024 elements (256 threads).
// mode 0: lower median (element 511). mode 1: true median (mean of 511,512).
// ---------------------------------------------------------------------------
__global__ __launch_bounds__(256) void col_select_kernel(
    const float* __restrict__ X, float* __restrict__ out, int mode) {
  __shared__ float sdat[TT];
  int m = blockIdx.x;
  for (int t = threadIdx.x; t < TT; t += blockDim.x) sdat[t] = X[(size_t)t * MM + m];
  __syncthreads();
  for (int ksz = 2; ksz <= TT; ksz <<= 1) {
    for (int j = ksz >> 1; j > 0; j >>= 1) {
      for (int l = threadIdx.x; l < TT; l += blockDim.x) {
        int p = l ^ j;
        if (p > l) {
          bool up = ((l & ksz) == 0);
          float a = sdat[l], b = sdat[p];
          if ((a > b) == up) { sdat[l] = b; sdat[p] = a; }
        }
      }
      __syncthreads();
    }
  }
  if (threadIdx.x == 0) {
    out[m] = mode ? 0.5f * (sdat[TT / 2 - 1] + sdat[TT / 2]) : sdat[(TT - 1) / 2];
  }
}

__global__ __launch_bounds__(256) void update_delta_kernel(
    const float* __restrict__ y, const float* __restrict__ yhat,
    const float* __restrict__ scol, float* __restrict__ delta) {
  int idx = blockIdx.x * blockDim.x + threadIdx.x;
  if (idx >= TT * MM) return;
  int m = idx % MM;
  float e = y[idx] - yhat[idx];
  float s6 = fmaxf(6.0f * scol[m], 1e-12f);
  float uu = fminf(fmaxf(e / s6, -1.0f), 1.0f);
  float t1 = 1.0f - uu * uu;
  delta[idx] = t1 * t1;
}

// Constant-column detection (block per column, OR-reduction)
__global__ __launch_bounds__(256) void const_check_kernel(
    const float* __restrict__ y, float* __restrict__ cflag) {
  __shared__ int sb[256];
  int m = blockIdx.x;
  float y0 = y[m];
  float tol = 1e-8f + 1e-5f * fabsf(y0);
  int bad = 0;
  for (int t = threadIdx.x; t < TT; t += blockDim.x)
    if (fabsf(y[(size_t)t * MM + m] - y0) > tol) bad = 1;
  sb[threadIdx.x] = bad;
  __syncthreads();
  for (int s = 128; s > 0; s >>= 1) {
    if (threadIdx.x < s) sb[threadIdx.x] |= sb[threadIdx.x + s];
    __syncthreads();
  }
  if (threadIdx.x == 0) cflag[m] = sb[0] ? 0.0f : 1.0f;
}

// trend = const ? y : yhat ; detrended = safe_div(y, trend); emit trend (B,T,C)
__global__ __launch_bounds__(256) void trend_detrend_kernel(
    const float* __restrict__ y, const float* __restrict__ yhat,
    const float* __restrict__ cflag, float* __restrict__ trend_tm,
    float* __restrict__ detr, float* __restrict__ out_trend) {
  int idx = blockIdx.x * blockDim.x + threadIdx.x;
  if (idx >= TT * MM) return;
  int t = idx / MM, m = idx % MM;
  float tr = (cflag[m] > 0.5f) ? y[idx] : yhat[idx];
  trend_tm[idx] = tr;
  float den = (fabsf(tr) < 1e-4f) ? 1e-4f : tr;
  detr[idx] = y[idx] / den;
  int b = m / CC, c = m % CC;
  out_trend[((size_t)b * TT + t) * CC + c] = tr;
}

// Per-(phase, column) means of detrended
__global__ __launch_bounds__(256) void seasonal_kernel(
    const float* __restrict__ detr, float* __restrict__ seas) {
  int idx = blockIdx.x * blockDim.x + threadIdx.x;
  if (idx >= PERIOD * MM) return;
  int p = idx / MM, m = idx % MM;
  float sum = 0.0f;
  int cnt = 0;
  for (int t = p; t < TT; t += PERIOD) { sum += detr[(size_t)t * MM + m]; cnt++; }
  seas[idx] = sum / (float)cnt;
}

// residual = safe_div(detrended, seasonal[t%12]); emit seasonal (B,T,C)
__global__ __launch_bounds__(256) void residual_kernel(
    const float* __restrict__ detr, const float* __restrict__ seas,
    float* __restrict__ resid, float* __restrict__ out_seasonal) {
  int idx = blockIdx.x * blockDim.x + threadIdx.x;
  if (idx >= TT * MM) return;
  int t = idx / MM, m = idx % MM;
  float se = seas[(t % PERIOD) * MM + m];
  float den = (fabsf(se) < 1e-4f) ? 1e-4f : se;
  resid[idx] = detr[idx] / den;
  int b = m / CC, c = m % CC;
  out_seasonal[((size_t)b * TT + t) * CC + c] = se;
}

__global__ __launch_bounds__(256) void absdev_kernel(
    const float* __restrict__ resid, const float* __restrict__ center,
    float* __restrict__ tmp) {
  int idx = blockIdx.x * blockDim.x + threadIdx.x;
  if (idx >= TT * MM) return;
  int m = idx % MM;
  tmp[idx] = fabsf(resid[idx] - center[m]);
}

__global__ __launch_bounds__(256) void finalize_kernel(
    const float* __restrict__ resid, const float* __restrict__ center,
    const float* __restrict__ madraw, float* __restrict__ out_anom,
    float* __restrict__ out_clean, float* __restrict__ out_mask) {
  int idx = blockIdx.x * blockDim.x + threadIdx.x;
  if (idx >= TT * MM) return;
  int t = idx / MM, m = idx % MM;
  float r = resid[idx];
  float ad = fabsf(r - center[m]);
  float mad = fmaxf(madraw[m], 1e-4f);
  float rob = 0.6745f * ad / mad;
  bool mask = rob > 3.5f;
  int b = m / CC, c = m % CC;
  size_t o = ((size_t)b * TT + t) * CC + c;
  out_anom[o]  = mask ? r : 1.0f;
  out_clean[o] = mask ? 1.0f : r;
  out_mask[o]  = mask ? 1.0f : 0.0f;
}

// ---------------------------------------------------------------------------
extern "C" void kernel_launch(void* const* d_in, const int* in_sizes, int n_in,
                              void* d_out, int out_size, void* d_ws, size_t ws_size,
                              hipStream_t stream) {
  const float* in = (const float*)d_in[0];
  float* out = (float*)d_out;
  float* ws = (float*)d_ws;

  const size_t TM = (size_t)TT * MM;   // 1M elements, also = B*T*C
  float* W     = ws;                   // TT x TT
  float* Y     = ws + TM;              // TT x MM
  float* DELTA = ws + 2 * TM;          // TT x MM
  float* R5    = ws + 3 * TM;          // TT x 5*MM
  float* S5    = ws + 8 * TM;          // TT x 5*MM
  float* YHAT  = ws + 13 * TM;         // TT x MM
  float* TMP   = ws + 14 * TM;         // TT x MM
  float* RESID = R5;                   // reuse R5 region after the fit loop
  float* TRTM  = R5 + TM;
  float* DETR  = DELTA;                // reuse delta after the fit loop
  float* SMALLS = ws + 15 * TM;
  float* SCOL   = SMALLS;              // MM
  float* CFLAG  = SMALLS + MM;         // MM
  float* CENTER = SMALLS + 2 * MM;     // MM
  float* MAD    = SMALLS + 3 * MM;     // MM
  float* SEAS   = SMALLS + 4 * MM;     // PERIOD x MM

  const int ew = (TT * MM + 255) / 256;               // elementwise grid
  dim3 gemmGrid(NN / 64, TT / 128);                   // (80, 8)

  build_w_kernel<<<(TT * TT + 255) / 256, 256, 0, stream>>>(W);
  transpose_init_kernel<<<ew, 256, 0, stream>>>(in, Y, DELTA);

  for (int it = 0; it < 4; ++it) {
    if (it > 0) {
      abs_err_kernel<<<ew, 256, 0, stream>>>(Y, YHAT, TMP);
      col_select_kernel<<<MM, 256, 0, stream>>>(TMP, SCOL, 1);   // true median
      update_delta_kernel<<<ew, 256, 0, stream>>>(Y, YHAT, SCOL, DELTA);
    }
    build_r5_kernel<<<ew, 256, 0, stream>>>(Y, DELTA, R5);
    wmma_gemm_f32_kernel<<<gemmGrid, 256, 0, stream>>>(W, R5, S5, TT, NN, TT);
    compute_yhat_kernel<<<ew, 256, 0, stream>>>(S5, YHAT);
  }

  const_check_kernel<<<MM, 256, 0, stream>>>(Y, CFLAG);
  trend_detrend_kernel<<<ew, 256, 0, stream>>>(Y, YHAT, CFLAG, TRTM, DETR, out);
  seasonal_kernel<<<(PERIOD * MM + 255) / 256, 256, 0, stream>>>(DETR, SEAS);
  residual_kernel<<<ew, 256, 0, stream>>>(DETR, SEAS, RESID, out + TM);
  col_select_kernel<<<MM, 256, 0, stream>>>(RESID, CENTER, 0);   // lower median
  absdev_kernel<<<ew, 256, 0, stream>>>(RESID, CENTER, TMP);
  col_select_kernel<<<MM, 256, 0, stream>>>(TMP, MAD, 0);        // lower median
  finalize_kernel<<<ew, 256, 0, stream>>>(RESID, CENTER, MAD,
                                          out + 2 * TM, out + 3 * TM, out + 4 * TM);
}